// TorchBandpassFIR_4844723110643
// MI455X (gfx1250) — compile-verified
//
#include <hip/hip_runtime.h>

typedef __attribute__((ext_vector_type(2))) float v2f;
typedef __attribute__((ext_vector_type(8))) float v8f;

#define K_TAPS 4096
#define PAD    2048
#define T_LEN  131072
#define S_SIG  32                     // B*C = 8*4 signals
#define TP     (T_LEN + 2*PAD)        // 135168 padded length
#define GT     2112                   // zero guard rows (time) each side
#define ROWS   (TP + 2*GT)            // 139392 (even)
#define GUARD  16                     // front guard inside h_ext
#define HEXT   (K_TAPS + 64)          // 4160 floats (zero-padded filter)
#define KWIN   (K_TAPS + 15)          // contraction window per 16-row tile
#define NCHUNK ((KWIN + 4) / 4)       // 1028 k-chunks of 4
#define TBASE  (-2048)                // even time base for BOTH passes

// ---------------------------------------------------------------------------
// Extended filters. Forward pass absorbs the odd 'same'-pad shift (-2047)
// into the filter: hext[i] = h[i - GUARD - 1], so A[m][k] = h[k - m - 1]
// with time base -2048. Backward pass uses the time-reversed filter
// unshifted: hrev[i] = h[K-1 - (i - GUARD)], time base -2048.
__global__ void prep_h_kernel(const float* __restrict__ h,
                              float* __restrict__ hext,
                              float* __restrict__ hrev) {
    int i = blockIdx.x * blockDim.x + threadIdx.x;
    if (i >= HEXT) return;
    int j = i - GUARD - 1;   // forward (shifted by 1)
    int d = i - GUARD;       // backward
    hext[i] = (j >= 0 && j < K_TAPS) ? h[j] : 0.f;
    hrev[i] = (d >= 0 && d < K_TAPS) ? h[K_TAPS - 1 - d] : 0.f;
}

// ---------------------------------------------------------------------------
// Staging layout: pair-interleaved transpose.  xpt[p][s][e] with
// p = row/2, e = row&1:  linear index = (row>>1)*64 + s*2 + (row&1).
// row = GT + padded-time; guards zero => branch-free WMMA loop.
// Threads map 1:1 onto the OUTPUT layout => fully coalesced stores.
__global__ void build_xpt_kernel(const float* __restrict__ x,
                                 float* __restrict__ xpt) {
    int o = blockIdx.x * blockDim.x + threadIdx.x;   // < ROWS*S_SIG
    int rem = o & 63;
    int s = rem >> 1;
    int t = ((o >> 6) << 1) + (rem & 1);             // row
    int tg = t - GT;                                 // padded-time coordinate
    float v = 0.f;
    if (tg >= 0 && tg < TP) {
        int tr = tg - PAD;                           // signal-time coordinate
        if (tr < 0)            tr = -tr;             // reflect left
        else if (tr >= T_LEN)  tr = 2 * T_LEN - 2 - tr; // reflect right
        v = x[(size_t)s * T_LEN + tr];
    }
    xpt[o] = v;
}

// Zero the guard rows of the intermediate (linear ranges are identical in
// the pair-interleaved layout; interior is fully overwritten by pass 1).
__global__ void zero_guards_kernel(float* __restrict__ y1t) {
    int idx = blockIdx.x * blockDim.x + threadIdx.x;   // < GT*S_SIG
    y1t[idx] = 0.f;
    y1t[(size_t)(GT + TP) * S_SIG + idx] = 0.f;
}

// ---------------------------------------------------------------------------
// One FIR pass as a Toeplitz GEMM on V_WMMA_F32_16X16X4_F32.
//   D[m][n] = sum_k hX[GUARD + k - m] * src[row = GT + t0 + TBASE + k][n']
// Each wave computes BOTH 16-signal column tiles (n' = n and n+16): one
// shared A fragment (ds_load_2addr) feeds two WMMAs whose B fragments are
// the two halves of the same 256B pair row (global_load_b64 x2).
// A fragment (16x4 f32): lanes 0-15 hold K=0,1 (vgpr0,1); lanes 16-31 K=2,3.
// C/D layout: vgpr r <-> row M=r (lanes 0-15) / M=r+8 (lanes 16-31).
// 8 waves per block sweep 8 consecutive time tiles in lockstep so the WGP$
// absorbs the ~8x window overlap before it reaches L2.
template<bool STORE_OUT>
__global__ __launch_bounds__(256)
void fir_pass_kernel(const float* __restrict__ hext,
                     const float* __restrict__ src,
                     float* __restrict__ dst,
                     int t0_origin) {
    __shared__ float sh[HEXT];
    for (int i = threadIdx.x; i < HEXT; i += 256) sh[i] = hext[i];
    __syncthreads();

    const int lane = threadIdx.x & 31;
    const int wave = threadIdx.x >> 5;
    const int n    = lane & 15;      // column (signal); also row m for A
    const int half = lane >> 4;      // 0: K=0,1   1: K=2,3
    const int q    = half * 2;

    const int t0 = t0_origin + (blockIdx.x * 8 + wave) * 16;

    int ab = GUARD + q - n;          // per-lane Toeplitz base into sh (>= 1)
    // first B time for this lane: GT + t0 + TBASE + q  (even)
    const float* p = src + ((size_t)(GT + t0 + TBASE + q) >> 1) * 64 + 2 * n;

    v8f acc0 = {};                   // columns 0..15
    v8f acc1 = {};                   // columns 16..31
#pragma unroll 4
    for (int kc = 0; kc < NCHUNK; ++kc) {
        v2f a, b0, b1;
        a.x = sh[ab];
        a.y = sh[ab + 1];
        b0 = *(const v2f*)p;         // times (t, t+1), columns n
        b1 = *(const v2f*)(p + 32);  // times (t, t+1), columns n+16
        acc0 = __builtin_amdgcn_wmma_f32_16x16x4_f32(
            false, a, false, b0, (short)0, acc0, false, false);
        acc1 = __builtin_amdgcn_wmma_f32_16x16x4_f32(
            false, a, false, b1, (short)0, acc1, false, false);
        ab += 4;
        p  += 128;                   // +4 time rows = +2 pairs
    }

    if (!STORE_OUT) {
        // intermediate in the same pair-interleaved layout; rows M=half*8+r
        const int row0 = GT + t0 + half * 8;           // even
        float* o = dst + ((size_t)row0 >> 1) * 64 + (size_t)n * 2;
#pragma unroll
        for (int r = 0; r < 4; ++r) {
            v2f w0, w1;
            w0.x = acc0[2 * r]; w0.y = acc0[2 * r + 1];
            w1.x = acc1[2 * r]; w1.y = acc1[2 * r + 1];
            *(v2f*)(o + (size_t)r * 64)      = w0;     // cols n
            *(v2f*)(o + (size_t)r * 64 + 32) = w1;     // cols n+16
        }
    } else {
        // final output, [signal][time]; each lane owns 8 consecutive times
        float* o0 = dst + (size_t)n        * T_LEN + (t0 - PAD) + half * 8;
        float* o1 = dst + (size_t)(n + 16) * T_LEN + (t0 - PAD) + half * 8;
        *(float4*)(o0)     = make_float4(acc0[0], acc0[1], acc0[2], acc0[3]);
        *((float4*)o0 + 1) = make_float4(acc0[4], acc0[5], acc0[6], acc0[7]);
        *(float4*)(o1)     = make_float4(acc1[0], acc1[1], acc1[2], acc1[3]);
        *((float4*)o1 + 1) = make_float4(acc1[4], acc1[5], acc1[6], acc1[7]);
    }
}

// ---------------------------------------------------------------------------
extern "C" void kernel_launch(void* const* d_in, const int* in_sizes, int n_in,
                              void* d_out, int out_size, void* d_ws, size_t ws_size,
                              hipStream_t stream) {
    const float* x    = (const float*)d_in[0];   // [8,4,131072] = [32][T]
    const float* kern = (const float*)d_in[1];   // [4096]
    float* out = (float*)d_out;
    float* ws  = (float*)d_ws;

    float* hext = ws;
    float* hrev = ws + HEXT;
    float* xpt  = ws + 2 * HEXT;                       // pair-interleaved
    float* y1t  = xpt + (size_t)ROWS * S_SIG;          // pair-interleaved

    prep_h_kernel<<<(HEXT + 255) / 256, 256, 0, stream>>>(kern, hext, hrev);
    build_xpt_kernel<<<(ROWS * S_SIG) / 256, 256, 0, stream>>>(x, xpt);
    zero_guards_kernel<<<(GT * S_SIG) / 256, 256, 0, stream>>>(y1t);

    // forward pass: y1 = corr(xp, h), 'same' left pad folded into hext
    fir_pass_kernel<false>
        <<<TP / 128, 256, 0, stream>>>(hext, xpt, y1t, 0);
    // backward pass (zero-phase), cropped to [PAD, PAD+T)
    fir_pass_kernel<true>
        <<<T_LEN / 128, 256, 0, stream>>>(hrev, y1t, out, PAD);
}